// GeomLossBarycentricPooling_77154792505937
// MI455X (gfx1250) — compile-verified
//
#include <hip/hip_runtime.h>

// GeomLoss barycentric pooling (batched log-domain Sinkhorn) for MI455X/gfx1250.
//
// Design:
//  * One workgroup (256 thr = 8 wave32) per graph b; 100 iterations fully fused
//    in a single kernel (graphs are independent -> only workgroup barriers).
//  * Cost matrix C[NS,32] built once with V_WMMA_F32_16X16X4_F32 (f32 needed:
//    eps=0.01 amplifies C errors x100 through exp; bf16 would be garbage).
//    0.5|x|^2 and 0.5|y|^2 are folded in via an augmented K=68 GEMM so the
//    WMMA accumulator *is* C. C (20.5MB) stays resident in the 192MB L2 and is
//    re-streamed 2x per iteration (~4.1GB of L2 traffic total).
//  * g-pass: lane = codebook column (coalesced 128B/row), 8 wave-groups stride
//    points, online (max,sum) LSE combined in LDS.
//  * f-pass: one point per wave, shfl_xor butterfly LSE over 32 columns.

typedef float v2f __attribute__((ext_vector_type(2)));
typedef float v8f __attribute__((ext_vector_type(8)));

namespace {
constexpr int   NNODES  = 40000;
constexpr int   SSAMP   = 4;
constexpr int   D       = 64;
constexpr int   K       = 32;
constexpr int   NB      = 128;
constexpr int   NITER   = 100;
constexpr int   NS      = NNODES * SSAMP;
constexpr float EPS     = 0.01f;     // blur^p = 0.1^2
constexpr float INV_EPS = 100.0f;
constexpr float NEG_BIG = -3.4028234e38f;
}

__device__ __forceinline__ int lb_search(const int* __restrict__ a, int n, int v) {
  int lo = 0, hi = n;
  while (lo < hi) { int mid = (lo + hi) >> 1; if (a[mid] < v) lo = mid + 1; else hi = mid; }
  return lo;
}

extern "C" __global__ __launch_bounds__(256, 1)
void sinkhorn_pool_kernel(const float* __restrict__ X,     // [NS, D]
                          const int*   __restrict__ bidx,  // [NNODES] sorted
                          const float* __restrict__ Y,     // [K, D]
                          float*       __restrict__ out,   // [NB, K]
                          float*       __restrict__ Cws,   // [NS, K] workspace
                          float*       __restrict__ fws)   // [NS]    workspace
{
  const int b    = blockIdx.x;
  const int tid  = threadIdx.x;
  const int lane = tid & 31;
  const int wave = tid >> 5;

  __shared__ float yyL[K];
  __shared__ float gL[K];
  __shared__ float redm[8][K];
  __shared__ float reds[8][K];

  // ---- segment range for this graph (sorted batch_idx -> binary search) ----
  const int nlo = lb_search(bidx, NNODES, b);
  const int nhi = lb_search(bidx, NNODES, b + 1);
  const int p0  = nlo * SSAMP;
  const int cnt = (nhi - nlo) * SSAMP;

  if (cnt == 0) {                      // empty graph -> uniform weights
    if (tid < K) out[b * K + tid] = 1.0f / (float)K;
    return;
  }
  const float log_a = -__logf((float)cnt);
  const float log_b = -__logf((float)K);

  // ---- 0.5*|y_j|^2 ----
  if (tid < K) {
    float s = 0.f;
    #pragma unroll
    for (int k = 0; k < D; ++k) { float y = Y[tid * D + k]; s += y * y; }
    yyL[tid] = 0.5f * s;
  }
  __syncthreads();

  // ---- Phase B: C = 0.5|x|^2 - x.y + 0.5|y|^2 via f32 WMMA, augmented K ----
  {
    const int   mrow  = lane & 15;      // row of tile (A) / col of tile (B,D)
    const int   khalf = lane >> 4;      // which half of each 4-wide k chunk
    const float yy0   = yyL[mrow];
    const float yy1   = yyL[16 + mrow];
    const int   ntiles = (cnt + 15) >> 4;

    for (int rt = wave; rt < ntiles; rt += 8) {
      const int rloc = rt * 16 + mrow;
      const int gp   = p0 + (rloc < cnt ? rloc : (cnt - 1));   // clamp pad rows
      const float* __restrict__ xr = X + (long)gp * D;

      v8f acc0 = {};  // cols 0..15
      v8f acc1 = {};  // cols 16..31
      float sq = 0.f;

      #pragma unroll
      for (int kk = 0; kk < 16; ++kk) {
        const int k = kk * 4 + khalf * 2;    // lane's 2 k-elements of chunk
        v2f a;  a.x = xr[k];                 a.y = xr[k + 1];
        sq += a.x * a.x + a.y * a.y;
        v2f b0; b0.x = -Y[mrow * D + k];        b0.y = -Y[mrow * D + k + 1];
        v2f b1; b1.x = -Y[(16 + mrow) * D + k]; b1.y = -Y[(16 + mrow) * D + k + 1];
        acc0 = __builtin_amdgcn_wmma_f32_16x16x4_f32(false, a, false, b0,
                                                     (short)0, acc0, false, false);
        acc1 = __builtin_amdgcn_wmma_f32_16x16x4_f32(false, a, false, b1,
                                                     (short)0, acc1, false, false);
      }
      // augmented chunk k=64..67: A = [0.5|x|^2, 1, 0, 0], B rows = [1s, 0.5|y|^2, 0, 0]
      const float sqf = sq + __shfl_xor(sq, 16, 32);   // full row norm
      v2f a, b0, b1;
      a.x  = (khalf == 0) ? 0.5f * sqf : 0.f;  a.y  = (khalf == 0) ? 1.0f : 0.f;
      b0.x = (khalf == 0) ? 1.0f       : 0.f;  b0.y = (khalf == 0) ? yy0  : 0.f;
      b1.x = (khalf == 0) ? 1.0f       : 0.f;  b1.y = (khalf == 0) ? yy1  : 0.f;
      acc0 = __builtin_amdgcn_wmma_f32_16x16x4_f32(false, a, false, b0,
                                                   (short)0, acc0, false, false);
      acc1 = __builtin_amdgcn_wmma_f32_16x16x4_f32(false, a, false, b1,
                                                   (short)0, acc1, false, false);

      // D layout: row = v + 8*(lane/16), col = lane%16 (+16 for acc1)
      #pragma unroll
      for (int v = 0; v < 8; ++v) {
        const int row = rt * 16 + v + 8 * khalf;
        if (row < cnt) {
          float* crow = Cws + (long)(p0 + row) * K;
          crow[mrow]      = acc0[v];
          crow[16 + mrow] = acc1[v];
        }
      }
    }
  }

  // ---- init f = 0 ----
  for (int p = tid; p < cnt; p += 256) fws[p0 + p] = 0.f;
  __threadfence();        // make C and f visible past the read-only WGP$
  __syncthreads();

  const float* __restrict__ Cseg = Cws + (long)p0 * K;
  float*       __restrict__ fseg = fws + p0;
  const int j   = lane;   // column owned in g-pass
  const int grp = wave;   // point-stride group in g-pass

  // ---- Sinkhorn iterations ----
  for (int it = 0; it < NITER; ++it) {
    // g[b,j] = -eps * LSE_i ((f_i - C_ij)/eps + log_a)   (online max/sum LSE)
    float m = NEG_BIG, s = 0.f;
    for (int p = grp; p < cnt; p += 8) {
      const float t = (fseg[p] - Cseg[p * K + j]) * INV_EPS + log_a;
      if (t > m) { s = s * __expf(m - t) + 1.0f; m = t; }
      else       { s += __expf(t - m); }
    }
    redm[grp][j] = m;
    reds[grp][j] = s;
    __syncthreads();
    if (tid < K) {
      float M = redm[0][tid];
      #pragma unroll
      for (int g2 = 1; g2 < 8; ++g2) M = fmaxf(M, redm[g2][tid]);
      float Ss = 0.f;
      #pragma unroll
      for (int g2 = 0; g2 < 8; ++g2) Ss += reds[g2][tid] * __expf(redm[g2][tid] - M);
      gL[tid] = -EPS * (M + __logf(fmaxf(Ss, 1e-30f)));
    }
    __syncthreads();

    // f_i = -eps * LSE_j ((g_j - C_ij)/eps + log_b) : one point per wave
    const float gj = gL[lane];
    for (int p = wave; p < cnt; p += 8) {
      const float t = (gj - Cseg[p * K + lane]) * INV_EPS + log_b;
      float mw = t;
      #pragma unroll
      for (int o = 16; o >= 1; o >>= 1) mw = fmaxf(mw, __shfl_xor(mw, o, 32));
      float e = __expf(t - mw);
      #pragma unroll
      for (int o = 16; o >= 1; o >>= 1) e += __shfl_xor(e, o, 32);
      if (lane == 0) fseg[p] = -EPS * (mw + __logf(e));
    }
    __threadfence();
    __syncthreads();
  }

  // ---- transport-plan column sums, normalized per graph ----
  {
    const float gj = gL[j];
    float acc = 0.f;
    for (int p = grp; p < cnt; p += 8) {
      const float lp = (fseg[p] + gj - Cseg[p * K + j]) * INV_EPS + log_a + log_b;
      acc += __expf(lp);
    }
    reds[grp][j] = acc;
    __syncthreads();
    if (tid < K) {   // tids 0..31 == one full wave
      float w = 0.f;
      #pragma unroll
      for (int g2 = 0; g2 < 8; ++g2) w += reds[g2][tid];
      float tot = w;
      #pragma unroll
      for (int o = 16; o >= 1; o >>= 1) tot += __shfl_xor(tot, o, 32);
      out[b * K + tid] = (tot > 0.f) ? (w / fmaxf(tot, 1e-30f)) : (1.0f / (float)K);
    }
  }
}

extern "C" void kernel_launch(void* const* d_in, const int* in_sizes, int n_in,
                              void* d_out, int out_size, void* d_ws, size_t ws_size,
                              hipStream_t stream) {
  (void)in_sizes; (void)n_in; (void)out_size; (void)ws_size;
  const float* X    = (const float*)d_in[0];   // [NS, D] fp32
  const int*   bidx = (const int*)  d_in[1];   // [NNODES] int32 sorted
  const float* Y    = (const float*)d_in[2];   // [K, D] fp32
  float*       o    = (float*)d_out;           // [NB, K] fp32

  float* Cws = (float*)d_ws;                   // NS*K floats = 20.48 MB
  float* fws = Cws + (size_t)NS * K;           // NS floats   = 0.64 MB

  sinkhorn_pool_kernel<<<dim3(NB), dim3(256), 0, stream>>>(X, bidx, Y, o, Cws, fws);
}